// NEQUIPLayer_63934883168898
// MI455X (gfx1250) — compile-verified
//
#include <hip/hip_runtime.h>
#include <math.h>

#define NNODE 10000
#define NEDGE 160000
#define C_CH 64
#define NS_CH 128
#define NV_CH 192

typedef __attribute__((ext_vector_type(2))) float v2f;
typedef __attribute__((ext_vector_type(8))) float v8f;

__device__ __forceinline__ v8f wmma4(v2f a, v2f b, v8f c) {
  // V_WMMA_F32_16X16X4_F32 : D = A(16x4 f32) * B(4x16 f32) + C(16x16 f32)
  return __builtin_amdgcn_wmma_f32_16x16x4_f32(false, a, false, b, (short)0, c, false, false);
}

__device__ __forceinline__ v8f vzero8() {
  v8f z;
#pragma unroll
  for (int j = 0; j < 8; ++j) z[j] = 0.0f;
  return z;
}

__device__ __forceinline__ float swishf(float x) {
  return x / (1.0f + __expf(-x));
}

// ---------------------------------------------------------------- zero scratch
__global__ void k_zero(float* __restrict__ p, int count) {
  for (int i = blockIdx.x * blockDim.x + threadIdx.x; i < count;
       i += gridDim.x * blockDim.x)
    p[i] = 0.0f;
}

// ---------------------------------------------------------------- linear_up
// one wave handles 16 nodes; WMMA f32 16x16x4, K=64
__global__ void k_up(const float* __restrict__ node_s,
                     const float* __restrict__ node_v,
                     const float* __restrict__ Wus,   // [64,64]
                     const float* __restrict__ Wuv,   // [64,64]
                     float* __restrict__ s_up,        // [N,64]
                     float* __restrict__ v_up) {      // [N,64,3]
  const int lane = threadIdx.x & 31;
  const int row  = lane & 15;          // A row (node) / B col (channel)
  const int bk   = (lane >> 4) << 1;   // 0 or 2 : K sub-index
  const int hi8  = (lane >> 4) << 3;   // D row offset
  const int node0 = blockIdx.x * 16;

  // ---- s_up = node_s @ W_up_s / 8
  v8f acc[4];
#pragma unroll
  for (int t = 0; t < 4; ++t) acc[t] = vzero8();
  for (int ks = 0; ks < 16; ++ks) {
    const int k = ks * 4 + bk;
    v2f a;
    a.x = node_s[(node0 + row) * 64 + k];
    a.y = node_s[(node0 + row) * 64 + k + 1];
#pragma unroll
    for (int t = 0; t < 4; ++t) {
      v2f b;
      b.x = Wus[k * 64 + t * 16 + row];
      b.y = Wus[(k + 1) * 64 + t * 16 + row];
      acc[t] = wmma4(a, b, acc[t]);
    }
  }
#pragma unroll
  for (int t = 0; t < 4; ++t)
#pragma unroll
    for (int j = 0; j < 8; ++j)
      s_up[(node0 + j + hi8) * 64 + t * 16 + row] = acc[t][j] * 0.125f;

  // ---- v_up[i] = node_v[:,:,i] @ W_up_v / 8
  for (int i = 0; i < 3; ++i) {
    v8f vacc[4];
#pragma unroll
    for (int t = 0; t < 4; ++t) vacc[t] = vzero8();
    for (int ks = 0; ks < 16; ++ks) {
      const int k = ks * 4 + bk;
      v2f a;
      a.x = node_v[((node0 + row) * 64 + k) * 3 + i];
      a.y = node_v[((node0 + row) * 64 + k + 1) * 3 + i];
#pragma unroll
      for (int t = 0; t < 4; ++t) {
        v2f b;
        b.x = Wuv[k * 64 + t * 16 + row];
        b.y = Wuv[(k + 1) * 64 + t * 16 + row];
        vacc[t] = wmma4(a, b, vacc[t]);
      }
    }
#pragma unroll
    for (int t = 0; t < 4; ++t)
#pragma unroll
      for (int j = 0; j < 8; ++j)
        v_up[((node0 + j + hi8) * 64 + t * 16 + row) * 3 + i] =
            vacc[t][j] * 0.125f;
  }
}

// ---------------------------------------------------------------- species skip (VALU)
__global__ void k_skip(const float* __restrict__ node_s,
                       const float* __restrict__ node_v,
                       const int*   __restrict__ specie,
                       const float* __restrict__ Wss,   // [S,64,128]
                       const float* __restrict__ Wsv,   // [S,64,64]
                       float* __restrict__ skip_s,      // [N,128]
                       float* __restrict__ skip_v) {    // [N,64,3]
  const int total_s = NNODE * NS_CH;
  const int total_v = NNODE * C_CH * 3;
  for (int idx = blockIdx.x * blockDim.x + threadIdx.x;
       idx < total_s + total_v; idx += gridDim.x * blockDim.x) {
    if (idx < total_s) {
      const int n = idx >> 7, o = idx & 127;
      const float* w = Wss + (size_t)specie[n] * 64 * 128 + o;
      const float* x = node_s + (size_t)n * 64;
      float acc = 0.0f;
      for (int c = 0; c < 64; ++c) acc += x[c] * w[c * 128];
      skip_s[idx] = acc * 0.125f;
    } else {
      const int r = idx - total_s;
      const int i = r % 3, oc = (r / 3) & 63, n = r / 192;
      const float* w = Wsv + (size_t)specie[n] * 64 * 64 + oc;
      const float* x = node_v + (size_t)n * 192 + i;
      float acc = 0.0f;
      for (int c = 0; c < 64; ++c) acc += x[c * 3] * w[c * 64];
      skip_v[r] = acc * 0.125f;   // [n][oc][i]
    }
  }
}

// ---------------------------------------------------------------- fused edge kernel
// one wave per 16 edges: radial MLP (3x WMMA layers) + tensor products + scatter
__global__ void k_edges(const float* __restrict__ vectors,
                        const int*   __restrict__ senders,
                        const int*   __restrict__ receivers,
                        const float* __restrict__ W1,   // [8,64]
                        const float* __restrict__ W2,   // [64,64]
                        const float* __restrict__ W3,   // [64,320]
                        const float* __restrict__ s_up, // [N,64]
                        const float* __restrict__ v_up, // [N,64,3]
                        float* __restrict__ agg_s,      // [N,128]
                        float* __restrict__ agg_v) {    // [N,192,3]
  __shared__ float lY[16][3];
  __shared__ int   lsnd[16], lrcv[16];
  __shared__ float h1[16][64];
  __shared__ float h2[16][64];
  __shared__ float mixs[16][320];

  const int lane = threadIdx.x & 31;
  const int row  = lane & 15;
  const int bk   = (lane >> 4) << 1;
  const int hi8  = (lane >> 4) << 3;
  const int e_loc = row;
  const int e = blockIdx.x * 16 + e_loc;

  // radial features (both lane halves compute their edge redundantly)
  const float vx = vectors[e * 3 + 0];
  const float vy = vectors[e * 3 + 1];
  const float vz = vectors[e * 3 + 2];
  const float r  = sqrtf(vx * vx + vy * vy + vz * vz);
  const float ri = 1.0f / r;
  const float Yx = 1.7320508075688772f * vx * ri;
  const float Yy = 1.7320508075688772f * vy * ri;
  const float Yz = 1.7320508075688772f * vz * ri;
  if (lane < 16) {
    lY[e_loc][0] = Yx; lY[e_loc][1] = Yy; lY[e_loc][2] = Yz;
    lsnd[e_loc] = senders[e];
    lrcv[e_loc] = receivers[e];
  }
  float x2 = r * r, x6 = x2 * x2 * x2;
  float env = 1.0f - 28.0f * x6 + 48.0f * x6 * r - 21.0f * x6 * x2;
  if (r >= 1.0f) env = 0.0f;
  const float be = 1.4142135623730951f * ri * env;
  float radk[2][2];   // this lane's A-fragment values for the 2 K-steps
#pragma unroll
  for (int ks = 0; ks < 2; ++ks)
#pragma unroll
    for (int p = 0; p < 2; ++p) {
      const int k = ks * 4 + bk + p;            // bessel index n = k+1
      radk[ks][p] = be * __sinf(3.14159265358979f * (float)(k + 1) * r);
    }
  __syncthreads();

  // ---- layer 1: [16,8] @ [8,64], /sqrt(8), swish
  {
    v8f c1[4];
#pragma unroll
    for (int t = 0; t < 4; ++t) c1[t] = vzero8();
#pragma unroll
    for (int ks = 0; ks < 2; ++ks) {
      const int k = ks * 4 + bk;
      v2f a; a.x = radk[ks][0]; a.y = radk[ks][1];
#pragma unroll
      for (int t = 0; t < 4; ++t) {
        v2f b;
        b.x = W1[k * 64 + t * 16 + row];
        b.y = W1[(k + 1) * 64 + t * 16 + row];
        c1[t] = wmma4(a, b, c1[t]);
      }
    }
#pragma unroll
    for (int t = 0; t < 4; ++t)
#pragma unroll
      for (int j = 0; j < 8; ++j)
        h1[j + hi8][t * 16 + row] = swishf(c1[t][j] * 0.35355339059327f);
  }
  __syncthreads();

  // ---- layer 2: [16,64] @ [64,64], /8, swish
  {
    v8f c2[4];
#pragma unroll
    for (int t = 0; t < 4; ++t) c2[t] = vzero8();
    for (int ks = 0; ks < 16; ++ks) {
      const int k = ks * 4 + bk;
      v2f a; a.x = h1[row][k]; a.y = h1[row][k + 1];
#pragma unroll
      for (int t = 0; t < 4; ++t) {
        v2f b;
        b.x = W2[k * 64 + t * 16 + row];
        b.y = W2[(k + 1) * 64 + t * 16 + row];
        c2[t] = wmma4(a, b, c2[t]);
      }
    }
#pragma unroll
    for (int t = 0; t < 4; ++t)
#pragma unroll
      for (int j = 0; j < 8; ++j)
        h2[j + hi8][t * 16 + row] = swishf(c2[t][j] * 0.125f);
  }
  __syncthreads();

  // ---- layer 3: [16,64] @ [64,320], /8 -> mix  (5 groups of 4 N-tiles)
  for (int tg = 0; tg < 5; ++tg) {
    v8f c3[4];
#pragma unroll
    for (int t = 0; t < 4; ++t) c3[t] = vzero8();
    for (int ks = 0; ks < 16; ++ks) {
      const int k = ks * 4 + bk;
      v2f a; a.x = h2[row][k]; a.y = h2[row][k + 1];
#pragma unroll
      for (int t = 0; t < 4; ++t) {
        const int n0 = (tg * 4 + t) * 16;
        v2f b;
        b.x = W3[k * 320 + n0 + row];
        b.y = W3[(k + 1) * 320 + n0 + row];
        c3[t] = wmma4(a, b, c3[t]);
      }
    }
#pragma unroll
    for (int t = 0; t < 4; ++t)
#pragma unroll
      for (int j = 0; j < 8; ++j)
        mixs[j + hi8][(tg * 4 + t) * 16 + row] = c3[t][j] * 0.125f;
  }
  __syncthreads();

  // ---- tensor products + gating + scatter-add (1/sqrt(16) folded into down)
  const float IS3 = 0.5773502691896258f;  // 1/sqrt(3)
  const float IS2 = 0.7071067811865476f;  // 1/sqrt(2)
  for (int e2 = 0; e2 < 16; ++e2) {
    const int snd = lsnd[e2], rcv = lrcv[e2];
    const float Y0 = lY[e2][0], Y1v = lY[e2][1], Y2v = lY[e2][2];
    const float* su = s_up + (size_t)snd * 64;
    const float* vu = v_up + (size_t)snd * 192;
    float* as = agg_s + (size_t)rcv * 128;
    float* av = agg_v + (size_t)rcv * 576;
    // scalar messages: [ms | dot(mv,Y)/sqrt3] * mix[:, :128]
    for (int c = lane; c < 128; c += 32) {
      float val;
      if (c < 64) {
        val = su[c];
      } else {
        const int c2 = c - 64;
        val = (vu[c2 * 3] * Y0 + vu[c2 * 3 + 1] * Y1v + vu[c2 * 3 + 2] * Y2v) * IS3;
      }
      atomicAdd(&as[c], val * mixs[e2][c]);
    }
    // vector messages: [mv | ms*Y | cross(mv,Y)/sqrt2] * mix[:, 128:]
    for (int c = lane; c < 192; c += 32) {
      const float m = mixs[e2][128 + c];
      float o0, o1, o2;
      if (c < 64) {
        o0 = vu[c * 3]; o1 = vu[c * 3 + 1]; o2 = vu[c * 3 + 2];
      } else if (c < 128) {
        const float s = su[c - 64];
        o0 = s * Y0; o1 = s * Y1v; o2 = s * Y2v;
      } else {
        const int c2 = c - 128;
        const float a0 = vu[c2 * 3], a1 = vu[c2 * 3 + 1], a2 = vu[c2 * 3 + 2];
        o0 = (a1 * Y2v - a2 * Y1v) * IS2;
        o1 = (a2 * Y0  - a0 * Y2v) * IS2;
        o2 = (a0 * Y1v - a1 * Y0 ) * IS2;
      }
      atomicAdd(&av[c * 3 + 0], o0 * m);
      atomicAdd(&av[c * 3 + 1], o1 * m);
      atomicAdd(&av[c * 3 + 2], o2 * m);
    }
  }
}

// ---------------------------------------------------------------- linear_down + skip + gate
__global__ void k_down(const float* __restrict__ agg_s,  // [N,128]
                       const float* __restrict__ agg_v,  // [N,192,3]
                       const float* __restrict__ Wds,    // [128,128]
                       const float* __restrict__ Wdv,    // [192,64]
                       const float* __restrict__ skip_s, // [N,128]
                       const float* __restrict__ skip_v, // [N,64,3]
                       float* __restrict__ out) {        // [N,256]
  __shared__ float so[16][128];
  const int lane = threadIdx.x & 31;
  const int row  = lane & 15;
  const int bk   = (lane >> 4) << 1;
  const int hi8  = (lane >> 4) << 3;
  const int node0 = blockIdx.x * 16;
  const float SDOWN = 0.022097086912079613f;  // (1/4)/sqrt(128)
  const float VDOWN = 0.018042195912175807f;  // (1/4)/sqrt(192)

  // out_s = agg_s @ W_down_s * SDOWN + skip_s   (8 N-tiles, K=128)
  for (int tg = 0; tg < 2; ++tg) {
    v8f acc[4];
#pragma unroll
    for (int t = 0; t < 4; ++t) acc[t] = vzero8();
    for (int ks = 0; ks < 32; ++ks) {
      const int k = ks * 4 + bk;
      v2f a;
      a.x = agg_s[(node0 + row) * 128 + k];
      a.y = agg_s[(node0 + row) * 128 + k + 1];
#pragma unroll
      for (int t = 0; t < 4; ++t) {
        const int n0 = (tg * 4 + t) * 16;
        v2f b;
        b.x = Wds[k * 128 + n0 + row];
        b.y = Wds[(k + 1) * 128 + n0 + row];
        acc[t] = wmma4(a, b, acc[t]);
      }
    }
#pragma unroll
    for (int t = 0; t < 4; ++t)
#pragma unroll
      for (int j = 0; j < 8; ++j) {
        const int m = j + hi8, col = (tg * 4 + t) * 16 + row;
        so[m][col] = acc[t][j] * SDOWN + skip_s[(node0 + m) * 128 + col];
      }
  }
  __syncthreads();

  // scalar outputs: swish(out_s[:, :64])
  for (int idx = lane; idx < 16 * 64; idx += 32) {
    const int m = idx >> 6, c = idx & 63;
    out[(node0 + m) * 256 + c] = swishf(so[m][c]);
  }

  // vector outputs: (agg_v @ W_down_v * VDOWN + skip_v) * swish(out_s[:, 64:])
  for (int i = 0; i < 3; ++i) {
    v8f acc[4];
#pragma unroll
    for (int t = 0; t < 4; ++t) acc[t] = vzero8();
    for (int ks = 0; ks < 48; ++ks) {
      const int k = ks * 4 + bk;
      v2f a;
      a.x = agg_v[((node0 + row) * 192 + k) * 3 + i];
      a.y = agg_v[((node0 + row) * 192 + k + 1) * 3 + i];
#pragma unroll
      for (int t = 0; t < 4; ++t) {
        v2f b;
        b.x = Wdv[k * 64 + t * 16 + row];
        b.y = Wdv[(k + 1) * 64 + t * 16 + row];
        acc[t] = wmma4(a, b, acc[t]);
      }
    }
#pragma unroll
    for (int t = 0; t < 4; ++t)
#pragma unroll
      for (int j = 0; j < 8; ++j) {
        const int m = j + hi8, c = t * 16 + row;
        const float g = swishf(so[m][64 + c]);
        const float val = acc[t][j] * VDOWN + skip_v[(node0 + m) * 192 + c * 3 + i];
        out[(node0 + m) * 256 + 64 + c * 3 + i] = val * g;
      }
  }
}

// ---------------------------------------------------------------- launch
extern "C" void kernel_launch(void* const* d_in, const int* in_sizes, int n_in,
                              void* d_out, int out_size, void* d_ws, size_t ws_size,
                              hipStream_t stream) {
  const float* vectors   = (const float*)d_in[0];
  const float* node_s    = (const float*)d_in[1];
  const float* node_v    = (const float*)d_in[2];
  const int*   specie    = (const int*)  d_in[3];
  const int*   senders   = (const int*)  d_in[4];
  const int*   receivers = (const int*)  d_in[5];
  const float* Wss       = (const float*)d_in[6];
  const float* Wsv       = (const float*)d_in[7];
  const float* Wus       = (const float*)d_in[8];
  const float* Wuv       = (const float*)d_in[9];
  const float* W1        = (const float*)d_in[10];
  const float* W2        = (const float*)d_in[11];
  const float* W3        = (const float*)d_in[12];
  const float* Wds       = (const float*)d_in[13];
  const float* Wdv       = (const float*)d_in[14];
  float* out = (float*)d_out;

  float* ws    = (float*)d_ws;
  float* s_up   = ws;                               // N*64
  float* v_up   = s_up   + (size_t)NNODE * 64;      // N*192
  float* skip_s = v_up   + (size_t)NNODE * 192;     // N*128
  float* skip_v = skip_s + (size_t)NNODE * 128;     // N*192
  float* agg_s  = skip_v + (size_t)NNODE * 192;     // N*128
  float* agg_v  = agg_s  + (size_t)NNODE * 128;     // N*576

  k_zero<<<512, 256, 0, stream>>>(agg_s, NNODE * (128 + 576));
  k_up  <<<NNODE / 16, 32, 0, stream>>>(node_s, node_v, Wus, Wuv, s_up, v_up);
  k_skip<<<2048, 256, 0, stream>>>(node_s, node_v, specie, Wss, Wsv, skip_s, skip_v);
  k_edges<<<NEDGE / 16, 32, 0, stream>>>(vectors, senders, receivers,
                                         W1, W2, W3, s_up, v_up, agg_s, agg_v);
  k_down<<<NNODE / 16, 32, 0, stream>>>(agg_s, agg_v, Wds, Wdv,
                                        skip_s, skip_v, out);
}